// GAT_15917148799234
// MI455X (gfx1250) — compile-verified
//
#include <hip/hip_runtime.h>
#include <math.h>

// ---------------------------------------------------------------------------
// GAT (2-layer, PyG-style) for gfx1250. fp32 WMMA for the two GEMMs
// (memory/atomic-bound workload -> keep reference fp32 numerics).
// ---------------------------------------------------------------------------

typedef __attribute__((ext_vector_type(2))) float v2f;
typedef __attribute__((ext_vector_type(8))) float v8f;

#define NEG_SLOPE 0.2f

__device__ __forceinline__ float leaky(float v) {
    return (v >= 0.0f) ? v : NEG_SLOPE * v;
}

// Float atomic max via signed/unsigned integer ordering trick.
// Valid when slot is initialized to -inf (0xFF800000).
__device__ __forceinline__ void atomicMaxF(float* addr, float v) {
    if (v >= 0.0f) {
        atomicMax((int*)addr, __float_as_int(v));
    } else {
        atomicMin((unsigned int*)addr, __float_as_uint(v));
    }
}

// ---------------------------------------------------------------------------
// GEMM1: h1[N,256] = x[N,128] @ W1[128,256]  (fp32 WMMA 16x16x4)
// block = 256 threads = 8 waves; wave w covers columns [w*32, w*32+32).
// A frag (16x4 f32): lanes 0-15 -> K=k+0,k+1 ; lanes 16-31 -> K=k+2,k+3.
// ---------------------------------------------------------------------------
__global__ __launch_bounds__(256) void gat_gemm1(const float* __restrict__ x,
                                                 const float* __restrict__ W,
                                                 float* __restrict__ h1, int N) {
    const int lane  = threadIdx.x & 31;
    const int wave  = threadIdx.x >> 5;
    const int mBase = blockIdx.x * 16;
    const int nBase = wave * 32;
    const int kOff  = (lane >> 4) * 2;
    const int nCol  = lane & 15;

    int mRow = mBase + (lane & 15);
    if (mRow >= N) mRow = N - 1;                 // clamp (N is tile-aligned here)
    const float* xr = x + (size_t)mRow * 128;

    v8f c0 = {}; v8f c1 = {};
    for (int k = 0; k < 128; k += 4) {
        if (k + 64 < 128) __builtin_prefetch(xr + k + 64, 0, 0);
        v2f a;
        a.x = xr[k + kOff];
        a.y = xr[k + kOff + 1];
        const float* w0 = W + (size_t)(k + kOff) * 256;
        v2f b0, b1;
        b0.x = w0[nBase + nCol];
        b0.y = w0[256 + nBase + nCol];
        b1.x = w0[nBase + 16 + nCol];
        b1.y = w0[256 + nBase + 16 + nCol];
        c0 = __builtin_amdgcn_wmma_f32_16x16x4_f32(false, a, false, b0, (short)0, c0, false, false);
        c1 = __builtin_amdgcn_wmma_f32_16x16x4_f32(false, a, false, b1, (short)0, c1, false, false);
    }
    const int mOut = mBase + (lane >> 4) * 8;    // D: VGPR r -> row r (+8 for hi lanes)
    for (int r = 0; r < 8; ++r) {
        int m = mOut + r;
        if (m < N) {
            h1[(size_t)m * 256 + nBase + nCol]      = c0[r];
            h1[(size_t)m * 256 + nBase + 16 + nCol] = c1[r];
        }
    }
}

// ---------------------------------------------------------------------------
// GEMM2: h2[N,40] = hrelu[N,256] @ W2[256,40]  (40 cols padded to 48, guarded)
// block = 96 threads = 3 waves; wave w covers columns [w*16, w*16+16).
// ---------------------------------------------------------------------------
__global__ __launch_bounds__(96) void gat_gemm2(const float* __restrict__ hrelu,
                                                const float* __restrict__ W2,
                                                float* __restrict__ h2, int N) {
    const int lane  = threadIdx.x & 31;
    const int wave  = threadIdx.x >> 5;
    const int mBase = blockIdx.x * 16;
    const int kOff  = (lane >> 4) * 2;
    const int nCol  = wave * 16 + (lane & 15);

    int mRow = mBase + (lane & 15);
    if (mRow >= N) mRow = N - 1;
    const float* xr = hrelu + (size_t)mRow * 256;

    v8f c = {};
    for (int k = 0; k < 256; k += 4) {
        v2f a;
        a.x = xr[k + kOff];
        a.y = xr[k + kOff + 1];
        v2f b;
        b.x = (nCol < 40) ? W2[(size_t)(k + kOff) * 40 + nCol]     : 0.0f;
        b.y = (nCol < 40) ? W2[(size_t)(k + kOff + 1) * 40 + nCol] : 0.0f;
        c = __builtin_amdgcn_wmma_f32_16x16x4_f32(false, a, false, b, (short)0, c, false, false);
    }
    if (nCol < 40) {
        const int mOut = mBase + (lane >> 4) * 8;
        for (int r = 0; r < 8; ++r) {
            int m = mOut + r;
            if (m < N) h2[(size_t)m * 40 + nCol] = c[r];
        }
    }
}

// ---------------------------------------------------------------------------
// Per-(node,head) attention scores, layer 1 (8 heads x 32 ch)
// ---------------------------------------------------------------------------
__global__ void gat_scores1(const float* __restrict__ h1, const float* __restrict__ aS,
                            const float* __restrict__ aD, float* __restrict__ sS,
                            float* __restrict__ sD, int N) {
    int t = blockIdx.x * blockDim.x + threadIdx.x;   // t = n*8 + head
    if (t >= N * 8) return;
    int n = t >> 3, h = t & 7;
    const float* hp = h1 + (size_t)n * 256 + h * 32;
    const float* as = aS + h * 32;
    const float* ad = aD + h * 32;
    float ss = 0.f, sd = 0.f;
    for (int c = 0; c < 32; ++c) { float v = hp[c]; ss += v * as[c]; sd += v * ad[c]; }
    sS[t] = ss; sD[t] = sd;
}

// Per-node scores, layer 2 (1 head x 40 ch)
__global__ void gat_scores2(const float* __restrict__ h2, const float* __restrict__ aS,
                            const float* __restrict__ aD, float* __restrict__ sS,
                            float* __restrict__ sD, int N) {
    int n = blockIdx.x * blockDim.x + threadIdx.x;
    if (n >= N) return;
    const float* hp = h2 + (size_t)n * 40;
    float ss = 0.f, sd = 0.f;
    for (int c = 0; c < 40; ++c) { float v = hp[c]; ss += v * aS[c]; sd += v * aD[c]; }
    sS[n] = ss; sD[n] = sd;
}

// ---------------------------------------------------------------------------
__global__ void fill_f32(float* __restrict__ p, float v, size_t n) {
    size_t i = (size_t)blockIdx.x * blockDim.x + threadIdx.x;
    if (i < n) p[i] = v;
}

// ---------------------------------------------------------------------------
// Edge passes, layer 1 (self-loops are edges e in [E, E+N))
// ---------------------------------------------------------------------------
__global__ void gat_edge_max1(const int* __restrict__ ei, int E, int N,
                              const float* __restrict__ sS, const float* __restrict__ sD,
                              float* __restrict__ m1) {
    int t = blockIdx.x * blockDim.x + threadIdx.x;   // t = e*8 + head
    int ET8 = (E + N) * 8;
    if (t >= ET8) return;
    int e = t >> 3, h = t & 7;
    int s = (e < E) ? ei[e]     : (e - E);
    int d = (e < E) ? ei[E + e] : (e - E);
    float v = leaky(sS[s * 8 + h] + sD[d * 8 + h]);
    atomicMaxF(&m1[d * 8 + h], v);
}

__global__ void gat_edge_sum1(const int* __restrict__ ei, int E, int N,
                              const float* __restrict__ sS, const float* __restrict__ sD,
                              const float* __restrict__ m1, float* __restrict__ den1) {
    int t = blockIdx.x * blockDim.x + threadIdx.x;
    int ET8 = (E + N) * 8;
    if (t >= ET8) return;
    int e = t >> 3, h = t & 7;
    int s = (e < E) ? ei[e]     : (e - E);
    int d = (e < E) ? ei[E + e] : (e - E);
    float v = leaky(sS[s * 8 + h] + sD[d * 8 + h]);
    atomicAdd(&den1[d * 8 + h], __expf(v - m1[d * 8 + h]));
}

// wave-per-(edge,head): lane = channel
__global__ __launch_bounds__(256) void gat_edge_agg1(const int* __restrict__ ei, int E, int N,
        const float* __restrict__ sS, const float* __restrict__ sD,
        const float* __restrict__ m1, const float* __restrict__ den1,
        const float* __restrict__ h1, float* __restrict__ agg1) {
    const int lane = threadIdx.x & 31;
    long long task = (long long)blockIdx.x * 8 + (threadIdx.x >> 5);
    long long ET8 = (long long)(E + N) * 8;
    if (task >= ET8) return;
    int e = (int)(task >> 3), h = (int)(task & 7);
    int s = (e < E) ? ei[e]     : (e - E);
    int d = (e < E) ? ei[E + e] : (e - E);
    float v = leaky(sS[s * 8 + h] + sD[d * 8 + h]);
    float alpha = __expf(v - m1[d * 8 + h]) / den1[d * 8 + h];
    float val = h1[(size_t)s * 256 + h * 32 + lane] * alpha;
    atomicAdd(&agg1[(size_t)d * 256 + h * 32 + lane], val);
}

__global__ void gat_bias_relu(float* __restrict__ agg, const float* __restrict__ b,
                              size_t total) {
    size_t i = (size_t)blockIdx.x * blockDim.x + threadIdx.x;
    if (i >= total) return;
    float v = agg[i] + b[i & 255];
    agg[i] = (v > 0.f) ? v : 0.f;
}

// ---------------------------------------------------------------------------
// Edge passes, layer 2 (1 head, 40 channels)
// ---------------------------------------------------------------------------
__global__ void gat_edge_max2(const int* __restrict__ ei, int E, int N,
                              const float* __restrict__ sS, const float* __restrict__ sD,
                              float* __restrict__ m2) {
    int e = blockIdx.x * blockDim.x + threadIdx.x;
    if (e >= E + N) return;
    int s = (e < E) ? ei[e]     : (e - E);
    int d = (e < E) ? ei[E + e] : (e - E);
    atomicMaxF(&m2[d], leaky(sS[s] + sD[d]));
}

__global__ void gat_edge_sum2(const int* __restrict__ ei, int E, int N,
                              const float* __restrict__ sS, const float* __restrict__ sD,
                              const float* __restrict__ m2, float* __restrict__ den2) {
    int e = blockIdx.x * blockDim.x + threadIdx.x;
    if (e >= E + N) return;
    int s = (e < E) ? ei[e]     : (e - E);
    int d = (e < E) ? ei[E + e] : (e - E);
    float v = leaky(sS[s] + sD[d]);
    atomicAdd(&den2[d], __expf(v - m2[d]));
}

// wave-per-edge, lanes cover the 40 channels
__global__ __launch_bounds__(256) void gat_edge_agg2(const int* __restrict__ ei, int E, int N,
        const float* __restrict__ sS, const float* __restrict__ sD,
        const float* __restrict__ m2, const float* __restrict__ den2,
        const float* __restrict__ h2, float* __restrict__ agg2) {
    const int lane = threadIdx.x & 31;
    long long e = (long long)blockIdx.x * 8 + (threadIdx.x >> 5);
    if (e >= (long long)(E + N)) return;
    int s = (e < E) ? ei[e]     : (int)(e - E);
    int d = (e < E) ? ei[E + e] : (int)(e - E);
    float v = leaky(sS[s] + sD[d]);
    float alpha = __expf(v - m2[d]) / den2[d];
    atomicAdd(&agg2[(size_t)d * 40 + lane], h2[(size_t)s * 40 + lane] * alpha);
    if (lane < 8)
        atomicAdd(&agg2[(size_t)d * 40 + 32 + lane], h2[(size_t)s * 40 + 32 + lane] * alpha);
}

__global__ void gat_logsoftmax(const float* __restrict__ agg2, const float* __restrict__ b2,
                               float* __restrict__ out, int N) {
    int n = blockIdx.x * blockDim.x + threadIdx.x;
    if (n >= N) return;
    float row[40];
    float mx = -__builtin_inff();
    for (int c = 0; c < 40; ++c) {
        float v = agg2[(size_t)n * 40 + c] + b2[c];
        row[c] = v;
        mx = fmaxf(mx, v);
    }
    float sum = 0.f;
    for (int c = 0; c < 40; ++c) sum += __expf(row[c] - mx);
    float lse = mx + __logf(sum);
    for (int c = 0; c < 40; ++c) out[(size_t)n * 40 + c] = row[c] - lse;
}

// ---------------------------------------------------------------------------
extern "C" void kernel_launch(void* const* d_in, const int* in_sizes, int n_in,
                              void* d_out, int out_size, void* d_ws, size_t ws_size,
                              hipStream_t stream) {
    const float* x   = (const float*)d_in[0];
    const int*   ei  = (const int*)  d_in[1];   // [2,E]: src then dst
    const float* W1  = (const float*)d_in[2];
    const float* aS1 = (const float*)d_in[3];
    const float* aD1 = (const float*)d_in[4];
    const float* b1  = (const float*)d_in[5];   // (zeros; aggregation epilogue uses it)
    const float* W2  = (const float*)d_in[6];
    const float* aS2 = (const float*)d_in[7];
    const float* aD2 = (const float*)d_in[8];
    const float* b2  = (const float*)d_in[9];
    float* out = (float*)d_out;

    const int N = in_sizes[0] / 128;
    const int E = in_sizes[1] / 2;
    const int ET = E + N;                        // with self-loops
    (void)n_in; (void)out_size; (void)ws_size; (void)b1;

    // Workspace layout (floats). Layer-2 buffers alias the dead h1 region.
    float* ws    = (float*)d_ws;
    float* h1    = ws;                           // N*256
    float* agg1  = h1 + (size_t)N * 256;         // N*256
    float* sS1   = agg1 + (size_t)N * 256;       // N*8
    float* sD1   = sS1 + (size_t)N * 8;          // N*8
    float* m1    = sD1 + (size_t)N * 8;          // N*8
    float* den1  = m1 + (size_t)N * 8;           // N*8
    float* h2    = h1;                           // N*40   (aliases h1, safe after agg1)
    float* sS2   = h2 + (size_t)N * 40;          // N
    float* sD2   = sS2 + N;                      // N
    float* m2    = sD2 + N;                      // N
    float* den2  = m2 + N;                       // N
    float* agg2  = den2 + N;                     // N*40

    const float NEG_INF = -__builtin_inff();
    const int mt = (N + 15) / 16;

    // ---- Layer 1 ----
    gat_gemm1<<<mt, 256, 0, stream>>>(x, W1, h1, N);
    gat_scores1<<<(N * 8 + 255) / 256, 256, 0, stream>>>(h1, aS1, aD1, sS1, sD1, N);
    fill_f32<<<((size_t)N * 8 + 255) / 256, 256, 0, stream>>>(m1, NEG_INF, (size_t)N * 8);
    fill_f32<<<((size_t)N * 8 + 255) / 256, 256, 0, stream>>>(den1, 0.0f, (size_t)N * 8);
    fill_f32<<<((size_t)N * 256 + 255) / 256, 256, 0, stream>>>(agg1, 0.0f, (size_t)N * 256);
    gat_edge_max1<<<(ET * 8 + 255) / 256, 256, 0, stream>>>(ei, E, N, sS1, sD1, m1);
    gat_edge_sum1<<<(ET * 8 + 255) / 256, 256, 0, stream>>>(ei, E, N, sS1, sD1, m1, den1);
    gat_edge_agg1<<<((long long)ET * 8 + 7) / 8, 256, 0, stream>>>(ei, E, N, sS1, sD1, m1, den1, h1, agg1);
    gat_bias_relu<<<((size_t)N * 256 + 255) / 256, 256, 0, stream>>>(agg1, b1, (size_t)N * 256);

    // ---- Layer 2 ----
    gat_gemm2<<<mt, 96, 0, stream>>>(agg1, W2, h2, N);
    gat_scores2<<<(N + 255) / 256, 256, 0, stream>>>(h2, aS2, aD2, sS2, sD2, N);
    fill_f32<<<(N + 255) / 256, 256, 0, stream>>>(m2, NEG_INF, (size_t)N);
    fill_f32<<<(N + 255) / 256, 256, 0, stream>>>(den2, 0.0f, (size_t)N);
    fill_f32<<<((size_t)N * 40 + 255) / 256, 256, 0, stream>>>(agg2, 0.0f, (size_t)N * 40);
    gat_edge_max2<<<(ET + 255) / 256, 256, 0, stream>>>(ei, E, N, sS2, sD2, m2);
    gat_edge_sum2<<<(ET + 255) / 256, 256, 0, stream>>>(ei, E, N, sS2, sD2, m2, den2);
    gat_edge_agg2<<<((long long)ET + 7) / 8, 256, 0, stream>>>(ei, E, N, sS2, sD2, m2, den2, h2, agg2);
    gat_logsoftmax<<<(N + 255) / 256, 256, 0, stream>>>(agg2, b2, out, N);
}